// Wavenet_51994874085608
// MI455X (gfx1250) — compile-verified
//
#include <hip/hip_runtime.h>
#include <math.h>

// ---------------- problem constants ----------------
constexpr int KSZ   = 5;
constexpr int NL    = 30;
constexpr int CYC   = 10;
constexpr int BATCH = 8;
constexpr int T     = 262144;

// ---------------- tiling ----------------
constexpr int NT    = 1024;          // threads per block (32 waves, wave32)
constexpr int TT    = 16384;         // output tile (samples)
constexpr int HALO  = 12288;         // >= receptive field 12280
constexpr int EXT   = TT + HALO;     // 28672 extended region in LDS
constexpr int PER   = EXT / NT;      // 28 positions per thread
constexpr int TS    = HALO / NT;     // 12: stripes >= TS are tile positions
constexpr int TILES = T / TT;        // 16 tiles per batch row
constexpr int NBLK  = BATCH * TILES; // 128 blocks for kernel A
constexpr size_t SMEM_A = (size_t)2 * EXT * sizeof(float); // 229376 B double-buffered h

constexpr float MULAW_LOG = 5.545177444479562f;  // log1p(255) = ln(256)
constexpr float NEG_BIG   = -3.402823466e38f;

// d_ws float layout: [0 .. 2*NBLK) per-block (m, s) partials;
//                    [512 .. 512+2*BATCH) per-row (M, 1/S)
constexpr int WS_ROWSTATS = 512;

__global__ __launch_bounds__(NT, 1)
void wavenet_layers(const float* __restrict__ x,
                    const float* __restrict__ cw, const float* __restrict__ cb,
                    const float* __restrict__ fw, const float* __restrict__ fb,
                    const float* __restrict__ gw, const float* __restrict__ gb,
                    const float* __restrict__ rw, const float* __restrict__ rb,
                    const float* __restrict__ c1w, const float* __restrict__ c1b,
                    const float* __restrict__ c2w, const float* __restrict__ c2b,
                    float* __restrict__ out, float* __restrict__ ws)
{
    extern __shared__ float smem[];
    float* ha = smem;        // h buffer A [EXT]
    float* hb = smem + EXT;  // h buffer B [EXT] (also x staging)

    const int tid  = threadIdx.x;
    const int row  = blockIdx.x / TILES;
    const int tile = blockIdx.x % TILES;
    const int gofs = tile * TT - HALO;          // global index of region position 0
    const float* __restrict__ xrow = x + (size_t)row * T;

    // ---- stage x[gofs .. gofs+EXT) into hb via async global->LDS (ASYNCcnt path) ----
    const int pv = (gofs < 0) ? -gofs : 0;      // zero (causal pad) prefix; multiple of 4096
    for (int p = tid; p < pv; p += NT) hb[p] = 0.f;
    {
        const int n4 = (EXT - pv) >> 2;          // exact multiple of NT (4096 or 7168)
        for (int c = tid; c < n4; c += NT) {
            const int p = pv + (c << 2);
            unsigned lds_addr = (unsigned)(size_t)(&hb[p]);     // LDS byte offset (addr[31:0])
            const float* ga = xrow + gofs + p;                   // 16B-aligned
            asm volatile("global_load_async_to_lds_b128 %0, %1, off"
                         :: "v"(lds_addr), "v"(ga) : "memory");
        }
    }
    asm volatile("s_wait_asynccnt 0" ::: "memory");
    __syncthreads();

    // ---- initial causal conv (d=1): ha = conv5(hb) + b ----
    {
        const float w0 = cw[0], w1 = cw[1], w2 = cw[2], w3 = cw[3], w4 = cw[4];
        const float b0 = cb[0];
        for (int i = 0; i < PER; ++i) {
            const int p = i * NT + tid;
            float a = b0 + w0 * hb[p];
            if (p >= 4) {
                a += w1 * hb[p - 1] + w2 * hb[p - 2] + w3 * hb[p - 3] + w4 * hb[p - 4];
            } else {
                if (p >= 1) a += w1 * hb[p - 1];
                if (p >= 2) a += w2 * hb[p - 2];
                if (p >= 3) a += w3 * hb[p - 3];
            }
            ha[p] = a;
        }
    }
    __syncthreads();

    // ---- 30 gated residual layers, fully LDS-resident, one barrier per layer ----
    float skip[PER - TS];
    #pragma unroll
    for (int i = 0; i < PER - TS; ++i) skip[i] = 0.f;

    float* cur = ha;
    float* nxt = hb;
    for (int k = 0; k < NL; ++k) {
        const int d = 1 << (k % CYC);
        const float fA0 = fw[k*KSZ+0], fA1 = fw[k*KSZ+1], fA2 = fw[k*KSZ+2],
                    fA3 = fw[k*KSZ+3], fA4 = fw[k*KSZ+4];
        const float gA0 = gw[k*KSZ+0], gA1 = gw[k*KSZ+1], gA2 = gw[k*KSZ+2],
                    gA3 = gw[k*KSZ+3], gA4 = gw[k*KSZ+4];
        const float fbk = fb[k], gbk = gb[k], rwk = rw[k], rbk = rb[k];

        // stripes 0..1: may need zero-clamped taps (p < 4*d possible, 4*d <= 2048)
        #pragma unroll
        for (int i = 0; i < 2; ++i) {
            const int p = i * NT + tid;
            const float h0 = cur[p];
            const float v1 = (p >=     d) ? cur[p -     d] : 0.f;
            const float v2 = (p >= 2 * d) ? cur[p - 2 * d] : 0.f;
            const float v3 = (p >= 3 * d) ? cur[p - 3 * d] : 0.f;
            const float v4 = (p >= 4 * d) ? cur[p - 4 * d] : 0.f;
            const float fpre = fbk + fA0*h0 + fA1*v1 + fA2*v2 + fA3*v3 + fA4*v4;
            const float gpre = gbk + gA0*h0 + gA1*v1 + gA2*v2 + gA3*v3 + gA4*v4;
            const float o = tanhf(fpre) * __builtin_amdgcn_rcpf(1.f + __expf(-gpre));
            nxt[p] = rwk * o + rbk + h0;
        }
        // stripes 2..27: p >= 2048 >= 4*d always -> unguarded fast path
        #pragma unroll
        for (int i = 2; i < PER; ++i) {
            const int p = i * NT + tid;
            const float h0 = cur[p];
            const float v1 = cur[p -     d];
            const float v2 = cur[p - 2 * d];
            const float v3 = cur[p - 3 * d];
            const float v4 = cur[p - 4 * d];
            const float fpre = fbk + fA0*h0 + fA1*v1 + fA2*v2 + fA3*v3 + fA4*v4;
            const float gpre = gbk + gA0*h0 + gA1*v1 + gA2*v2 + gA3*v3 + gA4*v4;
            const float o = tanhf(fpre) * __builtin_amdgcn_rcpf(1.f + __expf(-gpre));
            nxt[p] = rwk * o + rbk + h0;
            if (i >= TS) skip[i - TS] += o;     // constant index after unroll
        }
        __syncthreads();
        float* t = cur; cur = nxt; nxt = t;
    }

    // ---- epilogue: 1x1 convs + mu-law expansion; write pre-softmax values ----
    const float k1w = c1w[0], k1b = c1b[0], k2w = c2w[0], k2b = c2b[0];
    float* __restrict__ orow = out + (size_t)row * T + (size_t)tile * TT;
    float val[PER - TS];
    float vm = NEG_BIG;
    #pragma unroll
    for (int i = TS; i < PER; ++i) {
        const int p = i * NT + tid;
        const float s  = skip[i - TS];
        const float o1 = k1w * fmaxf(s,  0.f) + k1b;
        const float o2 = k2w * fmaxf(o1, 0.f) + k2b;
        const float mg = expm1f(fabsf(o2) * MULAW_LOG) * (1.0f / 255.0f);
        const float v  = (o2 >= 0.f) ? mg : -mg;
        val[i - TS] = v;
        orow[p - HALO] = v;
        vm = fmaxf(vm, v);
    }
    float vs = 0.f;
    #pragma unroll
    for (int i = 0; i < PER - TS; ++i) vs += __expf(val[i] - vm);

    // ---- fused online-softmax block reduction (reuse LDS; h buffers dead) ----
    float* redm = smem;
    float* reds = smem + NT;
    redm[tid] = vm;
    reds[tid] = vs;
    __syncthreads();
    for (int s = NT / 2; s > 0; s >>= 1) {
        if (tid < s) {
            const float m1 = redm[tid], m2 = redm[tid + s];
            const float M  = fmaxf(m1, m2);
            reds[tid] = reds[tid] * __expf(m1 - M) + reds[tid + s] * __expf(m2 - M);
            redm[tid] = M;
        }
        __syncthreads();
    }
    if (tid == 0) {
        ws[2 * blockIdx.x]     = redm[0];
        ws[2 * blockIdx.x + 1] = reds[0];
    }
}

// ---- combine the 16 per-tile (m,s) partials of each row -> (M, 1/S) ----
__global__ __launch_bounds__(32, 1)
void combine_stats(float* __restrict__ ws)
{
    const int row  = blockIdx.x;
    const int lane = threadIdx.x;          // wave32
    float m = NEG_BIG, s = 0.f;
    if (lane < TILES) {
        m = ws[2 * (row * TILES + lane)];
        s = ws[2 * (row * TILES + lane) + 1];
    }
    #pragma unroll
    for (int off = 16; off > 0; off >>= 1) {
        const float m2 = __shfl_down(m, off);
        const float s2 = __shfl_down(s, off);
        const float M  = fmaxf(m, m2);
        s = s * __expf(m - M) + s2 * __expf(m2 - M);
        m = M;
    }
    if (lane == 0) {
        ws[WS_ROWSTATS + 2 * row]     = m;
        ws[WS_ROWSTATS + 2 * row + 1] = 1.0f / s;
    }
}

// ---- fully parallel normalize: out = exp(v - M) * invS, float4 streams ----
__global__ __launch_bounds__(NT, 1)
void normalize(float* __restrict__ out, const float* __restrict__ ws)
{
    const int i4   = blockIdx.x * NT + threadIdx.x;   // float4 index
    const int row  = (blockIdx.x * NT) >> 16;         // T/4 = 65536 float4 per row (block-uniform)
    const float M    = ws[WS_ROWSTATS + 2 * row];
    const float invS = ws[WS_ROWSTATS + 2 * row + 1];
    float4 v = ((const float4*)out)[i4];
    v.x = __expf(v.x - M) * invS;
    v.y = __expf(v.y - M) * invS;
    v.z = __expf(v.z - M) * invS;
    v.w = __expf(v.w - M) * invS;
    ((float4*)out)[i4] = v;
}

extern "C" void kernel_launch(void* const* d_in, const int* in_sizes, int n_in,
                              void* d_out, int out_size, void* d_ws, size_t ws_size,
                              hipStream_t stream)
{
    (void)in_sizes; (void)n_in; (void)ws_size; (void)out_size;
    const float* x   = (const float*)d_in[0];
    const float* cw  = (const float*)d_in[1];
    const float* cb  = (const float*)d_in[2];
    const float* fwp = (const float*)d_in[3];
    const float* fbp = (const float*)d_in[4];
    const float* gwp = (const float*)d_in[5];
    const float* gbp = (const float*)d_in[6];
    const float* rwp = (const float*)d_in[7];
    const float* rbp = (const float*)d_in[8];
    const float* c1w = (const float*)d_in[9];
    const float* c1b = (const float*)d_in[10];
    const float* c2w = (const float*)d_in[11];
    const float* c2b = (const float*)d_in[12];
    float* out = (float*)d_out;
    float* ws  = (float*)d_ws;

    hipFuncSetAttribute(reinterpret_cast<const void*>(wavenet_layers),
                        hipFuncAttributeMaxDynamicSharedMemorySize, (int)SMEM_A);

    wavenet_layers<<<NBLK, NT, SMEM_A, stream>>>(
        x, cw, cb, fwp, fbp, gwp, gbp, rwp, rbp, c1w, c1b, c2w, c2b, out, ws);
    combine_stats<<<BATCH, 32, 0, stream>>>(ws);
    normalize<<<(BATCH * T / 4) / NT, NT, 0, stream>>>(out, ws);
}